// Otuken3DLoss_33844342292668
// MI455X (gfx1250) — compile-verified
//
#include <hip/hip_runtime.h>
#include <math.h>

// ---------------------------------------------------------------------------
// Otuken3D mesh loss for MI455X (gfx1250, wave32).
//
// Chamfer term dominates the FLOPs: 4 batches of a 5000x5000 distance matrix
// built from K=3 dot products. We zero-pad K to 4 and use the CDNA5
// V_WMMA_F32_16X16X4_F32 instruction: one wave computes a 16x16 tile of
// p.q per WMMA op (1024 MACs/instr). Inputs are ~240KB -> fully L2 resident;
// this pass is compute/latency bound, so f32 WMMA is the right precision.
//
// This revision makes the column loop branch-free (clamped loads + cndmask
// +inf on padded columns instead of exec-divergent guards) and processes TWO
// row tiles per wave so each B-operand load feeds two WMMAs.
//
// A-matrix (16x4, rows = pred points) follows the ISA-documented layout:
//   lanes 0-15 : VGPR0=K0, VGPR1=K1 ; lanes 16-31 : VGPR0=K2, VGPR1=K3
// B-matrix (4x16, cols = target points) mirrors the C-matrix striping:
//   VGPR0: lanes0-15=K0, lanes16-31=K2 ; VGPR1: lanes0-15=K1, lanes16-31=K3
// (compile-only environment: B layout is the consistent mirror assumption;
//  A/B agree on K-slot pairing and slot K=3 is zero in both, so the reduced
//  sum over K is layout-order independent.)
//
// Edge / normal / Laplacian terms are tiny gather/scatter kernels with
// deterministic block-partial reductions (Laplacian scatter uses
// GLOBAL_ATOMIC_ADD_F32). Final single-block kernel sums partials in fixed
// order -> deterministic scalar.
// ---------------------------------------------------------------------------

#define BATCH 4
#define ROW_TILES 2   // row tiles (of 16) per wave

typedef float v2f __attribute__((ext_vector_type(2)));
typedef float v8f __attribute__((ext_vector_type(8)));

// -------------------- Chamfer: row-min over all columns --------------------
// One wave (32 threads) per (32-row block, batch). Loops over all column
// tiles; each iteration: one clamped global_load_b96 + q2 math feeds TWO
// v_wmma_f32_16x16x4_f32 ops. Branch-free steady state (EXEC stays all-ones).
__global__ __launch_bounds__(32) void chamfer_rowmin_kernel(
    const float* __restrict__ P,   // [B][NP][3] rows (min target)
    const float* __restrict__ Q,   // [B][NQ][3] cols (min over these)
    float* __restrict__ rowmin,    // [B][NP]
    int NP, int NQ)
{
    const int lane  = threadIdx.x;      // 0..31
    const int hi    = lane >> 4;        // 0: lanes 0-15, 1: lanes 16-31
    const int l16   = lane & 15;
    const int nBase = blockIdx.x * (16 * ROW_TILES);
    const int b     = blockIdx.y;

    const float* Pb = P + (size_t)b * NP * 3;
    const float* Qb = Q + (size_t)b * NQ * 3;

    const float INF = __int_as_float(0x7f800000);

    // ---- A tiles: 16x4 (M x K), K=3 slot zero-padded ----
    v2f a[ROW_TILES];
    float p2v[ROW_TILES][8];
    float rmin[ROW_TILES][8];
#pragma unroll
    for (int rt = 0; rt < ROW_TILES; ++rt) {
        const int tBase = nBase + rt * 16;
        {
            const int n = tBase + l16;
            const int nc = min(n, NP - 1);          // clamped, always valid
            float x = Pb[nc * 3 + 0], y = Pb[nc * 3 + 1], z = Pb[nc * 3 + 2];
            if (n >= NP) { x = 0.0f; y = 0.0f; z = 0.0f; }  // cndmask, no branch
            if (hi == 0) { a[rt].x = x; a[rt].y = y;    }
            else         { a[rt].x = z; a[rt].y = 0.0f; }
        }
        // |p|^2 for the 8 rows this lane's accumulators map to (C layout:
        // VGPR r -> M = r + 8*hi, all 16 N-lanes share it).
#pragma unroll
        for (int r = 0; r < 8; ++r) {
            const int n  = tBase + r + 8 * hi;
            const int nc = min(n, NP - 1);
            const float x = Pb[nc * 3 + 0], y = Pb[nc * 3 + 1], z = Pb[nc * 3 + 2];
            p2v[rt][r] = (n < NP) ? (x * x + y * y + z * z) : 0.0f;
            rmin[rt][r] = INF;
        }
    }

    const int nTiles = (NQ + 15) / 16;
    for (int t = 0; t < nTiles; ++t) {
        const int m  = t * 16 + l16;          // this lane's column
        const int mc = min(m, NQ - 1);        // clamped -> load always legal

        const float x = Qb[mc * 3 + 0], y = Qb[mc * 3 + 1], z = Qb[mc * 3 + 2];
        // +inf on padded columns: d = p2 + inf - 2*dot = inf, never the min.
        const float q2 = (m < NQ) ? (x * x + y * y + z * z) : INF;
        v2f bv;
        if (hi == 0) { bv.x = x; bv.y = y;    }
        else         { bv.x = z; bv.y = 0.0f; }

#pragma unroll
        for (int rt = 0; rt < ROW_TILES; ++rt) {
            // dot tile: acc[r] = p[tBase + r + 8*hi] . q[t*16 + l16]
            v8f acc = {};
            acc = __builtin_amdgcn_wmma_f32_16x16x4_f32(
                /*neg_a=*/false, a[rt], /*neg_b=*/false, bv,
                /*c_mod=*/(short)0, acc, /*reuse_a=*/false, /*reuse_b=*/false);
#pragma unroll
            for (int r = 0; r < 8; ++r) {
                float d = p2v[rt][r] + q2 - 2.0f * acc[r];
                d = fmaxf(d, 0.0f);
                rmin[rt][r] = fminf(rmin[rt][r], d);
            }
        }
    }

    // Min-reduce across the 16 N-lanes of each half (xor masks 8..1 stay
    // within a 16-lane group on wave32), then lane 0 of each half writes.
#pragma unroll
    for (int rt = 0; rt < ROW_TILES; ++rt) {
#pragma unroll
        for (int r = 0; r < 8; ++r) {
            float v = rmin[rt][r];
#pragma unroll
            for (int s = 8; s >= 1; s >>= 1)
                v = fminf(v, __shfl_xor(v, s, 32));
            rmin[rt][r] = v;
        }
        if (l16 == 0) {
#pragma unroll
            for (int r = 0; r < 8; ++r) {
                const int n = nBase + rt * 16 + r + 8 * hi;
                if (n < NP) rowmin[(size_t)b * NP + n] = rmin[rt][r];
            }
        }
    }
}

// -------------------- small helpers --------------------
__device__ __forceinline__ void block_reduce_store(float v, float* partial)
{
    __shared__ float sdata[256];
    sdata[threadIdx.x] = v;
    __syncthreads();
#pragma unroll
    for (int s = 128; s > 0; s >>= 1) {
        if ((int)threadIdx.x < s) sdata[threadIdx.x] += sdata[threadIdx.x + s];
        __syncthreads();
    }
    if (threadIdx.x == 0) partial[blockIdx.x] = sdata[0];
}

__global__ void zero_kernel(float* p, int n)
{
    const int i = blockIdx.x * 256 + threadIdx.x;
    if (i < n) p[i] = 0.0f;
}

// -------------------- edge loss --------------------
__global__ __launch_bounds__(256) void edge_loss_kernel(
    const float* __restrict__ verts, const int* __restrict__ edges,
    int E, int Vc, float* __restrict__ partial)
{
    const int i = blockIdx.x * 256 + threadIdx.x;
    float val = 0.0f;
    if (i < BATCH * E) {
        const int b = i / E, e = i % E;
        const int e0 = edges[e * 2 + 0], e1 = edges[e * 2 + 1];
        const float* vb = verts + (size_t)b * Vc * 3;
        const float dx = vb[e0 * 3 + 0] - vb[e1 * 3 + 0];
        const float dy = vb[e0 * 3 + 1] - vb[e1 * 3 + 1];
        const float dz = vb[e0 * 3 + 2] - vb[e1 * 3 + 2];
        val = dx * dx + dy * dy + dz * dz;
    }
    block_reduce_store(val, partial);
}

// -------------------- normal consistency loss --------------------
__device__ __forceinline__ void face_normal(const float* vb, const int* faces,
                                            int f, float n[3])
{
    const int v0 = faces[f * 3 + 0], v1 = faces[f * 3 + 1], v2 = faces[f * 3 + 2];
    const float ax = vb[v1 * 3 + 0] - vb[v0 * 3 + 0];
    const float ay = vb[v1 * 3 + 1] - vb[v0 * 3 + 1];
    const float az = vb[v1 * 3 + 2] - vb[v0 * 3 + 2];
    const float bx = vb[v2 * 3 + 0] - vb[v0 * 3 + 0];
    const float by = vb[v2 * 3 + 1] - vb[v0 * 3 + 1];
    const float bz = vb[v2 * 3 + 2] - vb[v0 * 3 + 2];
    n[0] = ay * bz - az * by;
    n[1] = az * bx - ax * bz;
    n[2] = ax * by - ay * bx;
    const float inv = 1.0f / (sqrtf(n[0] * n[0] + n[1] * n[1] + n[2] * n[2]) + 1e-8f);
    n[0] *= inv; n[1] *= inv; n[2] *= inv;
}

__global__ __launch_bounds__(256) void normal_loss_kernel(
    const float* __restrict__ verts, const int* __restrict__ faces,
    const int* __restrict__ pairs, int P, int Vc, float* __restrict__ partial)
{
    const int i = blockIdx.x * 256 + threadIdx.x;
    float val = 0.0f;
    if (i < BATCH * P) {
        const int b = i / P, pe = i % P;
        const float* vb = verts + (size_t)b * Vc * 3;
        float n0[3], n1[3];
        face_normal(vb, faces, pairs[pe * 2 + 0], n0);
        face_normal(vb, faces, pairs[pe * 2 + 1], n1);
        val = 1.0f - (n0[0] * n1[0] + n0[1] * n1[1] + n0[2] * n1[2]);
    }
    block_reduce_store(val, partial);
}

// -------------------- Laplacian loss --------------------
__global__ __launch_bounds__(256) void lap_scatter_kernel(
    const float* __restrict__ verts, const int* __restrict__ edges,
    int E, int Vc, float* __restrict__ nb, float* __restrict__ deg)
{
    const int i = blockIdx.x * 256 + threadIdx.x;
    if (i >= BATCH * E) return;
    const int b = i / E, e = i % E;
    const int e0 = edges[e * 2 + 0], e1 = edges[e * 2 + 1];
    const float* vb = verts + (size_t)b * Vc * 3;
    float* nbb = nb + (size_t)b * Vc * 3;
#pragma unroll
    for (int c = 0; c < 3; ++c) {
        atomicAdd(&nbb[e0 * 3 + c], vb[e1 * 3 + c]);  // global_atomic_add_f32
        atomicAdd(&nbb[e1 * 3 + c], vb[e0 * 3 + c]);
    }
    if (b == 0) {
        atomicAdd(&deg[e0], 1.0f);
        atomicAdd(&deg[e1], 1.0f);
    }
}

__global__ __launch_bounds__(256) void lap_loss_kernel(
    const float* __restrict__ verts, const float* __restrict__ nb,
    const float* __restrict__ deg, int Vc, float* __restrict__ partial)
{
    const int i = blockIdx.x * 256 + threadIdx.x;
    float val = 0.0f;
    if (i < BATCH * Vc) {
        const int b = i / Vc, v = i % Vc;
        const float dg = fmaxf(deg[v], 1.0f);
        float s = 0.0f;
#pragma unroll
        for (int c = 0; c < 3; ++c) {
            const size_t idx = ((size_t)b * Vc + v) * 3 + c;
            const float l = nb[idx] / dg - verts[idx];
            s += l * l;
        }
        val = sqrtf(s + 1e-12f);
    }
    block_reduce_store(val, partial);
}

// -------------------- final deterministic reduction --------------------
__device__ float block_sum_256(const float* __restrict__ p, int n, float* sdata)
{
    float acc = 0.0f;
    for (int i = threadIdx.x; i < n; i += 256) acc += p[i];
    sdata[threadIdx.x] = acc;
    __syncthreads();
#pragma unroll
    for (int s = 128; s > 0; s >>= 1) {
        if ((int)threadIdx.x < s) sdata[threadIdx.x] += sdata[threadIdx.x + s];
        __syncthreads();
    }
    const float r = sdata[0];
    __syncthreads();
    return r;
}

__global__ __launch_bounds__(256) void final_kernel(
    const float* __restrict__ minP, int nMinP,
    const float* __restrict__ minQ, int nMinQ,
    const float* __restrict__ partE, int nPartE, float divE,
    const float* __restrict__ partN, int nPartN, float divN,
    const float* __restrict__ partL, int nPartL, float divL,
    float* __restrict__ out)
{
    __shared__ float sdata[256];
    const float chamfer = block_sum_256(minP, nMinP, sdata) / (float)nMinP +
                          block_sum_256(minQ, nMinQ, sdata) / (float)nMinQ;
    const float edge    = block_sum_256(partE, nPartE, sdata) / divE;
    const float normal  = block_sum_256(partN, nPartN, sdata) / divN;
    const float lap     = block_sum_256(partL, nPartL, sdata) / divL;
    if (threadIdx.x == 0)
        out[0] = 1.0f * chamfer + 0.1f * edge + 0.01f * normal + 0.1f * lap;
}

// ---------------------------------------------------------------------------
extern "C" void kernel_launch(void* const* d_in, const int* in_sizes, int n_in,
                              void* d_out, int out_size, void* d_ws, size_t ws_size,
                              hipStream_t stream)
{
    const float* pred  = (const float*)d_in[0];
    const float* tgt   = (const float*)d_in[1];
    const float* verts = (const float*)d_in[2];
    const int*   faces = (const int*)d_in[3];
    const int*   edges = (const int*)d_in[4];
    const int*   pairs = (const int*)d_in[5];

    const int N  = in_sizes[0] / (BATCH * 3);   // 5000
    const int M  = in_sizes[1] / (BATCH * 3);   // 5000
    const int Vc = in_sizes[2] / (BATCH * 3);   // 16384
    const int E  = in_sizes[4] / 2;             // ~48641
    const int P  = in_sizes[5] / 2;             // ~48133

    const int gridE = (BATCH * E  + 255) / 256;
    const int gridP = (BATCH * P  + 255) / 256;
    const int gridL = (BATCH * Vc + 255) / 256;

    // workspace carve-up (floats); ~1.03 MB total
    float* ws   = (float*)d_ws;
    float* minP = ws;                size_t off = (size_t)BATCH * N;
    float* minQ = ws + off;          off += (size_t)BATCH * M;
    float* nb   = ws + off;          off += (size_t)BATCH * Vc * 3;
    float* deg  = ws + off;          off += (size_t)Vc;
    float* partE = ws + off;         off += (size_t)gridE;
    float* partN = ws + off;         off += (size_t)gridP;
    float* partL = ws + off;         off += (size_t)gridL;
    (void)ws_size; (void)n_in; (void)out_size;

    // Chamfer, both directions (WMMA tiles), no atomics -> deterministic.
    const int rowsPerWave = 16 * ROW_TILES;
    dim3 cgP((N + rowsPerWave - 1) / rowsPerWave, BATCH);
    chamfer_rowmin_kernel<<<cgP, 32, 0, stream>>>(pred, tgt, minP, N, M);
    dim3 cgQ((M + rowsPerWave - 1) / rowsPerWave, BATCH);
    chamfer_rowmin_kernel<<<cgQ, 32, 0, stream>>>(tgt, pred, minQ, M, N);

    // Laplacian: zero nb+deg (contiguous), scatter, then per-vertex loss.
    const int zeroCnt = BATCH * Vc * 3 + Vc;
    zero_kernel<<<(zeroCnt + 255) / 256, 256, 0, stream>>>(nb, zeroCnt);
    lap_scatter_kernel<<<(BATCH * E + 255) / 256, 256, 0, stream>>>(
        verts, edges, E, Vc, nb, deg);
    lap_loss_kernel<<<gridL, 256, 0, stream>>>(verts, nb, deg, Vc, partL);

    edge_loss_kernel<<<gridE, 256, 0, stream>>>(verts, edges, E, Vc, partE);
    normal_loss_kernel<<<gridP, 256, 0, stream>>>(verts, faces, pairs, P, Vc, partN);

    final_kernel<<<1, 256, 0, stream>>>(
        minP, BATCH * N, minQ, BATCH * M,
        partE, gridE, (float)(BATCH * E),
        partN, gridP, (float)(BATCH * P),
        partL, gridL, (float)(BATCH * Vc),
        (float*)d_out);
}